// VectorQuantizer_46308337386260
// MI455X (gfx1250) — compile-verified
//
#include <hip/hip_runtime.h>

typedef __attribute__((ext_vector_type(2))) float v2f;
typedef __attribute__((ext_vector_type(8))) float v8f;

#define NUM_CODES 8192
#define DIM       64
#define NROWS     32768          // 32 * 32 * 32 vectors
#define Q_ELEMS   2097152        // 32 * 64 * 32 * 32
#define BETA      0.25f

// ---------------------------------------------------------------------------
// Kernel 1: codebook squared norms -> ws[0..8191]
// ---------------------------------------------------------------------------
__global__ __launch_bounds__(256) void vq_cnorm_kernel(const float* __restrict__ cb,
                                                       float* __restrict__ cnorm) {
    int k = blockIdx.x * blockDim.x + threadIdx.x;
    if (k < NUM_CODES) {
        const float4* p = (const float4*)(cb + (long)k * DIM);
        float s = 0.f;
#pragma unroll
        for (int i = 0; i < 16; ++i) {
            float4 v = p[i];
            s += v.x * v.x + v.y * v.y + v.z * v.z + v.w * v.w;
        }
        cnorm[k] = s;
    }
}

// ---------------------------------------------------------------------------
// Kernel 2: WMMA distance scan + argmin + gather/scatter + loss partials.
// 256 blocks x 256 threads (8 waves). Each wave owns 16 rows of z_flat and
// scans all 8192 codes in 16-column tiles with V_WMMA_F32_16X16X4_F32.
// ---------------------------------------------------------------------------
__global__ __launch_bounds__(256) void vq_main_kernel(
    const float* __restrict__ z, const float* __restrict__ cb,
    const float* __restrict__ cnorm, float* __restrict__ qout,
    float* __restrict__ idxout, float* __restrict__ partial) {

    const int lane  = threadIdx.x & 31;
    const int wave  = threadIdx.x >> 5;
    const int tile  = blockIdx.x * 8 + wave;   // 0..2047
    const int row0  = tile * 16;

    const int half  = lane >> 4;               // 0: lanes 0-15, 1: lanes 16-31
    const int l16   = lane & 15;
    const int kbase = half * 2;                // A/B frag: V0 -> k=kbase, V1 -> k=kbase+1

    // ---- preload A fragments: rows row0..row0+15 of z_flat, all 64 dims ----
    // z_flat[r][d] = z[(r>>10)*65536 + d*1024 + (r & 1023)]  (NCHW source)
    const int  m    = row0 + l16;                              // this lane's A row
    const long zoff = (long)(m >> 10) * 65536 + (m & 1023);
    v2f a[16];
#pragma unroll
    for (int kk = 0; kk < 16; ++kk) {
        int d = kk * 4 + kbase;
        a[kk].x = z[zoff + (long)d * 1024];
        a[kk].y = z[zoff + (long)(d + 1) * 1024];
    }

    float minval[8];
    int   minidx[8];
#pragma unroll
    for (int v = 0; v < 8; ++v) { minval[v] = 3.4e38f; minidx[v] = 0; }

    // ---- scan all codes, 16 at a time ----
    for (int c0 = 0; c0 < NUM_CODES; c0 += 16) {
        // B[k][n] = cb[n*64 + k]; lane supplies col n = c0+l16, rows kbase/kbase+1
        const float* bp = cb + (long)(c0 + l16) * DIM + kbase;
        v8f acc = {};
#pragma unroll
        for (int kk = 0; kk < 16; ++kk) {
            v2f b = *(const v2f*)(bp + kk * 4);   // 8B-aligned global_load_b64
            acc = __builtin_amdgcn_wmma_f32_16x16x4_f32(
                false, a[kk], false, b, (short)0, acc, false, false);
        }
        // C layout: lane holds (row = v + 8*half, col = c0 + l16) in acc[v]
        float cn = cnorm[c0 + l16];
#pragma unroll
        for (int v = 0; v < 8; ++v) {
            float d2 = cn - 2.0f * acc[v];        // |z|^2 constant per row: dropped
            if (d2 < minval[v]) { minval[v] = d2; minidx[v] = c0 + l16; }
        }
    }

    // ---- min-reduce across each 16-lane half (xor 1,2,4,8 stays in half) ----
#pragma unroll
    for (int v = 0; v < 8; ++v) {
        float val = minval[v];
        int   idx = minidx[v];
#pragma unroll
        for (int off = 1; off < 16; off <<= 1) {
            float oval = __shfl_xor(val, off, 32);
            int   oidx = __shfl_xor(idx, off, 32);
            if (oval < val || (oval == val && oidx < idx)) { val = oval; idx = oidx; }
        }
        minval[v] = val;
        minidx[v] = idx;
    }

    // ---- gather codebook rows, scatter NCHW, accumulate (q-z)^2 ----
    float lsum = 0.f;
#pragma unroll
    for (int v = 0; v < 8; ++v) {
        int r  = row0 + v + half * 8;             // row handled by this half
        int ci = minidx[v];
        if (l16 == 0) idxout[r] = (float)ci;
        const float4 q4 = *(const float4*)(cb + (long)ci * DIM + l16 * 4);
        long base = (long)(r >> 10) * 65536 + (r & 1023);
        float qv[4] = {q4.x, q4.y, q4.z, q4.w};
#pragma unroll
        for (int j = 0; j < 4; ++j) {
            long off = base + (long)(l16 * 4 + j) * 1024;
            float df = qv[j] - z[off];
            lsum += df * df;
            qout[off] = qv[j];                    // straight-through == quantized
        }
    }

    // ---- block-level loss partial (fixed-order tree; deterministic) ----
    __shared__ float red[256];
    red[threadIdx.x] = lsum;
    __syncthreads();
    for (int s = 128; s > 0; s >>= 1) {
        if ((int)threadIdx.x < s) red[threadIdx.x] += red[threadIdx.x + s];
        __syncthreads();
    }
    if (threadIdx.x == 0) partial[blockIdx.x] = red[0];
}

// ---------------------------------------------------------------------------
// Kernel 3: reduce 256 partials -> vq_loss = (1+BETA) * mean((q-z)^2)
// ---------------------------------------------------------------------------
__global__ __launch_bounds__(256) void vq_loss_kernel(const float* __restrict__ partial,
                                                      float* __restrict__ loss_out) {
    __shared__ float red[256];
    red[threadIdx.x] = partial[threadIdx.x];
    __syncthreads();
    for (int s = 128; s > 0; s >>= 1) {
        if ((int)threadIdx.x < s) red[threadIdx.x] += red[threadIdx.x + s];
        __syncthreads();
    }
    if (threadIdx.x == 0)
        *loss_out = red[0] * (1.0f + BETA) / (float)Q_ELEMS;
}

extern "C" void kernel_launch(void* const* d_in, const int* in_sizes, int n_in,
                              void* d_out, int out_size, void* d_ws, size_t ws_size,
                              hipStream_t stream) {
    const float* z  = (const float*)d_in[0];   // [32, 64, 32, 32]
    const float* cb = (const float*)d_in[1];   // [8192, 64]

    float* out    = (float*)d_out;
    float* qout   = out;                 // 2097152 floats (NCHW quantized_st)
    float* loss   = out + Q_ELEMS;       // 1 float
    float* idxout = out + Q_ELEMS + 1;   // 32768 floats (indices)

    float* cnorm   = (float*)d_ws;            // 8192 floats
    float* partial = (float*)d_ws + NUM_CODES; // 256 floats

    vq_cnorm_kernel<<<NUM_CODES / 256, 256, 0, stream>>>(cb, cnorm);
    vq_main_kernel<<<256, 256, 0, stream>>>(z, cb, cnorm, qout, idxout, partial);
    vq_loss_kernel<<<1, 256, 0, stream>>>(partial, loss);
    (void)in_sizes; (void)n_in; (void)out_size; (void)ws_size;
}